// LaplacianLossBatch_29686813950546
// MI455X (gfx1250) — compile-verified
//
#include <hip/hip_runtime.h>

#ifndef __has_builtin
#define __has_builtin(x) 0
#endif

#define BB      16          // batches
#define NV      2562        // vertices per mesh
#define NF      5120        // faces per mesh
#define WORDS   81          // ceil(NV/32) bitmask words per adjacency row
#define RPB     432         // rows (vertices) handled per workgroup
#define CHUNKS  6           // ceil(NV/RPB): 6*432 = 2592 >= 2562
#define THREADS 512         // 16 wave32 waves
#define VTXF    (NV * 3)    // 7686 floats = 30744 B
#define VTXPAD  7688        // pad to 16B so adjacency region is uint4-aligned
#define ADJW    (RPB * WORDS)                   // 34992 words = 139968 B
#define SMEM_BYTES (VTXPAD * 4 + ADJW * 4 + 16) // ~170.7 KB (< 320 KB WGP LDS)

typedef __attribute__((address_space(1))) int gint_t;   // prints as "__device__ int"
typedef __attribute__((address_space(3))) int lint_t;   // prints as "__shared__ int"

__global__ void zero_out_kernel(float* out) {
    if (threadIdx.x == 0) out[0] = 0.0f;
}

__global__ __launch_bounds__(THREADS, 1) void lap_loss_kernel(
    const float* __restrict__ verts,   // (B, NV, 3) f32
    const int*   __restrict__ faces,   // (B, NF, 3) i32
    float*       __restrict__ out)     // scalar accumulator
{
    extern __shared__ char smem[];
    float*    vtx  = (float*)smem;                         // VTXF floats
    unsigned* adj  = (unsigned*)(smem + VTXPAD * 4);       // ADJW words
    float*    bsum = (float*)(smem + VTXPAD * 4 + ADJW * 4);

    const int b    = blockIdx.y;
    const int row0 = blockIdx.x * RPB;
    const int row1 = (row0 + RPB < NV) ? (row0 + RPB) : NV;
    const int tid  = threadIdx.x;

    const float* vsrc = verts + (size_t)b * VTXF;

    // ---- Stage the whole batch's vertex array into LDS (async-to-LDS path) ----
#if __has_builtin(__builtin_amdgcn_global_load_async_to_lds_b32)
#define ASYNC_STAGED 1
    for (int t = tid; t < VTXF; t += THREADS) {
        __builtin_amdgcn_global_load_async_to_lds_b32(
            (gint_t*)(vsrc + t),
            (lint_t*)(vtx + t),
            /*imm offset*/0, /*cpol*/0);
    }
#else
    for (int t = tid; t < VTXF; t += THREADS) vtx[t] = vsrc[t];
#endif

    // ---- Zero the adjacency bitset with 16B LDS stores (overlaps async loads) ----
    uint4* adj4 = (uint4*)adj;
    const uint4 z4 = make_uint4(0u, 0u, 0u, 0u);
    for (int t = tid; t < ADJW / 4; t += THREADS) adj4[t] = z4;
    if (tid == 0) *bsum = 0.0f;

#ifdef ASYNC_STAGED
#if __has_builtin(__builtin_amdgcn_s_wait_asynccnt)
    __builtin_amdgcn_s_wait_asynccnt(0);
#else
    asm volatile("s_wait_asynccnt 0" ::: "memory");
#endif
#endif
    __syncthreads();

    // ---- Insert edges: 6 directed pairs per face, idempotent ds_or_b32 ----
    const int* fsrc = faces + (size_t)b * (NF * 3);
    for (int f = tid; f < NF; f += THREADS) {
        const int i0 = fsrc[3 * f + 0];
        const int i1 = fsrc[3 * f + 1];
        const int i2 = fsrc[3 * f + 2];
#define SETDIR(rr, cc)                                                        \
        do {                                                                  \
            const int _r = (rr), _c = (cc);                                   \
            if (_r >= row0 && _r < row1)                                      \
                atomicOr(&adj[(_r - row0) * WORDS + (_c >> 5)],               \
                         1u << (_c & 31));                                    \
        } while (0)
        SETDIR(i0, i1); SETDIR(i1, i0);
        SETDIR(i1, i2); SETDIR(i2, i1);
        SETDIR(i2, i0); SETDIR(i0, i2);
#undef SETDIR
    }
    __syncthreads();

    // ---- One wave32 per row: popcount degree + gather neighbors from LDS ----
    const int lane = tid & 31;
    const int wave = tid >> 5;
    for (int r = row0 + wave; r < row1; r += THREADS / 32) {
        const unsigned* rowp  = adj + (size_t)(r - row0) * WORDS;
        const int       selfw = r >> 5;
        const unsigned  selfm = 1u << (r & 31);
        int   deg = 0;
        float sx = 0.0f, sy = 0.0f, sz = 0.0f;
        for (int w = lane; w < WORDS; w += 32) {
            unsigned m = rowp[w];
            deg += __popc(m);            // self-bit counts toward degree ...
            if (w == selfw) m &= ~selfm; // ... but is excluded from the sum
            while (m) {
                const int j = (w << 5) + (__ffs(m) - 1);
                m &= (m - 1);
                sx += vtx[3 * j + 0];
                sy += vtx[3 * j + 1];
                sz += vtx[3 * j + 2];
            }
        }
        // wave32 butterfly reduction
        for (int off = 16; off > 0; off >>= 1) {
            deg += __shfl_xor(deg, off, 32);
            sx  += __shfl_xor(sx,  off, 32);
            sy  += __shfl_xor(sy,  off, 32);
            sz  += __shfl_xor(sz,  off, 32);
        }
        if (lane == 0) {
            const float d     = (float)deg;
            const float denom = (deg == 0) ? 1.0f : d;
            const float ax = (d * vtx[3 * r + 0] - sx) / denom;
            const float ay = (d * vtx[3 * r + 1] - sy) / denom;
            const float az = (d * vtx[3 * r + 2] - sz) / denom;
            atomicAdd(bsum, ax * ax + ay * ay + az * az); // ds_add_f32
        }
    }
    __syncthreads();
    if (tid == 0) atomicAdd(out, *bsum); // global_atomic_add_f32
}

extern "C" void kernel_launch(void* const* d_in, const int* in_sizes, int n_in,
                              void* d_out, int out_size, void* d_ws, size_t ws_size,
                              hipStream_t stream) {
    (void)in_sizes; (void)n_in; (void)out_size; (void)d_ws; (void)ws_size;
    const float* verts = (const float*)d_in[0];
    const int*   faces = (const int*)d_in[1];
    float*       out   = (float*)d_out;

    zero_out_kernel<<<1, 32, 0, stream>>>(out);
    dim3 grid(CHUNKS, BB, 1);
    lap_loss_kernel<<<grid, THREADS, SMEM_BYTES, stream>>>(verts, faces, out);
}